// GAT_41927470743992
// MI455X (gfx1250) — compile-verified
//
#include <hip/hip_runtime.h>

// ---------------- problem constants (match reference) ----------------
#define NN      100000      // nodes
#define NE      1600000     // edges (before self loops)
#define NT      (NE + NN)   // edges incl. self loops
#define INF_    128         // in feats
#define HC      128         // heads*out (4*32)
#define NHEAD   4
#define COUT    32
#define EDIM    16
#define NEG_SLOPE 0.2f

typedef __attribute__((ext_vector_type(2))) float v2f;
typedef __attribute__((ext_vector_type(8))) float v8f;

// ======================================================================
// Setup: build augmented transposed weight WTa[144][128] and w_e_att[16][4]
//   cols 0..127   : W^T
//   cols 128..131 : v_src[k][h] = sum_c W[k][h*32+c]*att_src[h][c]
//   cols 132..135 : v_dst likewise
//   cols 136..143 : zero padding (to 9 column tiles of 16)
// ======================================================================
__global__ void gat_setup(const float* __restrict__ W, const float* __restrict__ We,
                          const float* __restrict__ att_src, const float* __restrict__ att_dst,
                          const float* __restrict__ att_edge,
                          float* __restrict__ WTa, float* __restrict__ w_e_att) {
    int tid = threadIdx.x;
    for (int i = tid; i < 128 * 128; i += 256) {
        int col = i >> 7, k = i & 127;
        WTa[col * 128 + k] = W[k * 128 + col];
    }
    for (int i = tid; i < 16 * 128; i += 256) {
        int cc = i >> 7, k = i & 127;   // cc in [0,16): augmented column
        float v = 0.f;
        if (cc < 4) {
            for (int c = 0; c < 32; ++c) v += W[k * 128 + cc * 32 + c] * att_src[cc * 32 + c];
        } else if (cc < 8) {
            int hh = cc - 4;
            for (int c = 0; c < 32; ++c) v += W[k * 128 + hh * 32 + c] * att_dst[hh * 32 + c];
        }
        WTa[(128 + cc) * 128 + k] = v;
    }
    if (tid < 64) {
        int d = tid >> 2, hh = tid & 3;
        float v = 0.f;
        for (int c = 0; c < 32; ++c) v += We[d * 128 + hh * 32 + c] * att_edge[hh * 32 + c];
        w_e_att[d * 4 + hh] = v;
    }
}

// ======================================================================
// Init: zero reduction buffers, seed d_out with bias (head-mean adds on top)
// ======================================================================
__global__ void gat_init(float* __restrict__ out, const float* __restrict__ bias,
                         float* __restrict__ ea_dot, float* __restrict__ cnt,
                         unsigned int* __restrict__ amax, float* __restrict__ denom) {
    int i = blockIdx.x * 256 + threadIdx.x;          // covers NN*32
    if (i < NN * 32) out[i] = bias[i & 31];
    if (i < NN * 4) { ea_dot[i] = 0.f; amax[i] = 0u; denom[i] = 0.f; }
    if (i < NN)     cnt[i] = 0.f;
}

// ======================================================================
// Fused GEMM: [N,128] x [128,144] via V_WMMA_F32_16X16X4_F32 (exact fp32).
// One wave -> 16 rows x 9 tiles (h cols 0..127, a_src 128..131, a_dst 132..135).
// K tiled 2x64 through LDS; LDS row stride 66 dwords (stride%64==2 -> no
// bank conflicts on the 16-lane column reads).
// ======================================================================
__launch_bounds__(256)
__global__ void gat_gemm(const float* __restrict__ x, const float* __restrict__ WTa,
                         float* __restrict__ h, float* __restrict__ a_src,
                         float* __restrict__ a_dst) {
    __shared__ float sW[144 * 66];                    // 38,016 B
    const int tid  = threadIdx.x;
    const int wave = tid >> 5;
    const int lane = tid & 31;
    const int lmod = lane & 15;                       // M (A) / N (B,C)
    const int lhi  = lane >> 4;                       // K half-select
    int rg = blockIdx.x * 8 + wave;                   // 16-row group
    const bool active = rg < (NN / 16);
    if (!active) rg = (NN / 16) - 1;                  // keep wave in barriers, full EXEC
    const int row0 = rg * 16;

    v8f acc[9];
    v8f z = {0.f,0.f,0.f,0.f,0.f,0.f,0.f,0.f};
    #pragma unroll
    for (int t = 0; t < 9; ++t) acc[t] = z;

    const float* xrow = x + (size_t)(row0 + lmod) * INF_;

    for (int kb2 = 0; kb2 < 128; kb2 += 64) {
        for (int i = tid; i < 144 * 64; i += 256) {   // stage K-half of WTa
            int col = i >> 6, kk = i & 63;
            sW[col * 66 + kk] = WTa[col * 128 + kb2 + kk];
        }
        __syncthreads();
        #pragma unroll
        for (int ks = 0; ks < 64; ks += 4) {
            const int klo = ks + lhi * 2;             // this lane's K pair
            v2f a = *(const v2f*)(xrow + kb2 + klo);  // A: lane=M, vgpr j=K+j
            #pragma unroll
            for (int t = 0; t < 9; ++t) {
                v2f b = *(const v2f*)(&sW[(t * 16 + lmod) * 66 + klo]);
                acc[t] = __builtin_amdgcn_wmma_f32_16x16x4_f32(
                    false, a, false, b, (short)0, acc[t], false, false);
            }
        }
        __syncthreads();
    }

    if (active) {
        #pragma unroll
        for (int t = 0; t < 8; ++t)
            #pragma unroll
            for (int v = 0; v < 8; ++v)               // C/D: vgpr v -> M = v+8*lhi
                h[(size_t)(row0 + v + 8 * lhi) * HC + t * 16 + lmod] = acc[t][v];
        if (lmod < 8) {                               // tile 8: attention coefficients
            float* p = (lmod < 4) ? (a_src + lmod) : (a_dst + (lmod - 4));
            #pragma unroll
            for (int v = 0; v < 8; ++v)
                p[(size_t)(row0 + v + 8 * lhi) * 4] = acc[8][v];
        }
    }
}

// ======================================================================
// Pass A: per-edge s_edge = ea . w_e_att ; accumulate segment sums for
// the self-loop 'mean' edge_attr (as dot products, never 16-wide).
// ======================================================================
__global__ void gat_edgeA(const long long* __restrict__ eidx, const float* __restrict__ eattr,
                          const float* __restrict__ w_e_att, float* __restrict__ alpha,
                          float* __restrict__ ea_dot, float* __restrict__ cnt) {
    int e = blockIdx.x * 256 + threadIdx.x;
    if (e >= NE) return;
    int dst = (int)eidx[NE + e];
    const float* ea = eattr + (size_t)e * EDIM;
    float s[4] = {0.f, 0.f, 0.f, 0.f};
    #pragma unroll
    for (int d = 0; d < EDIM; ++d) {
        float v = ea[d];                              // w_e_att: uniform addr -> s_loads
        s[0] += v * w_e_att[d * 4 + 0];
        s[1] += v * w_e_att[d * 4 + 1];
        s[2] += v * w_e_att[d * 4 + 2];
        s[3] += v * w_e_att[d * 4 + 3];
    }
    #pragma unroll
    for (int hh = 0; hh < 4; ++hh) {
        alpha[(size_t)e * 4 + hh] = s[hh];
        atomicAdd(&ea_dot[dst * 4 + hh], s[hh]);
    }
    atomicAdd(&cnt[dst], 1.0f);
}

// ======================================================================
// Pass B: raw alpha + leaky relu + segment max (ordered-uint atomicMax).
// Self-loop edges are e in [NE, NT).
// ======================================================================
__global__ void gat_edgeB(const long long* __restrict__ eidx, const float* __restrict__ a_src,
                          const float* __restrict__ a_dst, const float* __restrict__ ea_dot,
                          const float* __restrict__ cnt, float* __restrict__ alpha,
                          unsigned int* __restrict__ amax) {
    int e = blockIdx.x * 256 + threadIdx.x;
    if (e >= NT) return;
    int src, dst; float s[4];
    if (e < NE) {
        src = (int)eidx[e]; dst = (int)eidx[NE + e];
        #pragma unroll
        for (int hh = 0; hh < 4; ++hh) s[hh] = alpha[(size_t)e * 4 + hh];
    } else {
        int n = e - NE; src = n; dst = n;
        float invc = 1.0f / fmaxf(cnt[n], 1.0f);
        #pragma unroll
        for (int hh = 0; hh < 4; ++hh) s[hh] = ea_dot[n * 4 + hh] * invc;
    }
    #pragma unroll
    for (int hh = 0; hh < 4; ++hh) {
        float a = a_src[src * 4 + hh] + a_dst[dst * 4 + hh] + s[hh];
        a = a > 0.f ? a : NEG_SLOPE * a;
        alpha[(size_t)e * 4 + hh] = a;
        unsigned int u = __float_as_uint(a);
        u = (u & 0x80000000u) ? ~u : (u | 0x80000000u);   // order-preserving encode
        atomicMax(&amax[dst * 4 + hh], u);
    }
}

// ======================================================================
// Pass C: exp(alpha - amax[dst]) ; segment-sum denominator.
// ======================================================================
__global__ void gat_edgeC(const long long* __restrict__ eidx, const unsigned int* __restrict__ amax,
                          float* __restrict__ alpha, float* __restrict__ denom) {
    int e = blockIdx.x * 256 + threadIdx.x;
    if (e >= NT) return;
    int dst = (e < NE) ? (int)eidx[NE + e] : (e - NE);
    #pragma unroll
    for (int hh = 0; hh < 4; ++hh) {
        unsigned int u = amax[dst * 4 + hh];
        float m = (u & 0x80000000u) ? __uint_as_float(u & 0x7FFFFFFFu) : __uint_as_float(~u);
        float ex = __expf(alpha[(size_t)e * 4 + hh] - m);
        alpha[(size_t)e * 4 + hh] = ex;
        atomicAdd(&denom[dst * 4 + hh], ex);
    }
}

// ======================================================================
// Pass D: wave per edge; lane = output channel. Head-mean fused:
// out[dst,c] += 0.25 * sum_h (alpha/denom) * h[src, h*32+c].
// h (51MB) + out are L2-resident on MI455X (192MB L2).
// ======================================================================
__global__ void gat_edgeD(const long long* __restrict__ eidx, const float* __restrict__ alpha,
                          const float* __restrict__ denom, const float* __restrict__ h,
                          float* __restrict__ out) {
    int e = blockIdx.x * 8 + (threadIdx.x >> 5);
    int lane = threadIdx.x & 31;
    if (e >= NT) return;
    int src, dst;
    if (e < NE) { src = (int)eidx[e]; dst = (int)eidx[NE + e]; }
    else        { src = e - NE; dst = src; }
    float s = 0.f;
    #pragma unroll
    for (int hh = 0; hh < 4; ++hh) {
        float coef = alpha[(size_t)e * 4 + hh] / denom[dst * 4 + hh];
        s += coef * h[(size_t)src * HC + hh * 32 + lane];
    }
    atomicAdd(&out[dst * 32 + lane], 0.25f * s);
}

// ======================================================================
extern "C" void kernel_launch(void* const* d_in, const int* in_sizes, int n_in,
                              void* d_out, int out_size, void* d_ws, size_t ws_size,
                              hipStream_t stream) {
    const float*     x        = (const float*)d_in[0];
    const long long* eidx     = (const long long*)d_in[1];   // int64 per reference
    const float*     eattr    = (const float*)d_in[2];
    const float*     W        = (const float*)d_in[3];
    const float*     We       = (const float*)d_in[4];
    const float*     att_src  = (const float*)d_in[5];
    const float*     att_dst  = (const float*)d_in[6];
    const float*     att_edge = (const float*)d_in[7];
    const float*     bias     = (const float*)d_in[8];
    float* out = (float*)d_out;

    // ---- workspace carve-up (all sizes 256B-aligned) ----
    char* ws = (char*)d_ws;
    size_t off = 0;
    float*        WTa     = (float*)(ws + off); off += 144 * 128 * 4;   // 73,728
    float*        w_e_att = (float*)(ws + off); off += 256;             // 16x4
    float*        a_src   = (float*)(ws + off); off += (size_t)NN * 4 * 4;
    float*        a_dst   = (float*)(ws + off); off += (size_t)NN * 4 * 4;
    float*        ea_dot  = (float*)(ws + off); off += (size_t)NN * 4 * 4;
    float*        cnt     = (float*)(ws + off); off += 400128;          // NN floats, padded
    unsigned int* amax    = (unsigned int*)(ws + off); off += (size_t)NN * 4 * 4;
    float*        denom   = (float*)(ws + off); off += (size_t)NN * 4 * 4;
    float*        h       = (float*)(ws + off); off += (size_t)NN * HC * 4;   // 51.2 MB
    float*        alpha   = (float*)(ws + off); off += (size_t)NT * 4 * 4;    // 27.2 MB
    (void)ws_size; (void)in_sizes; (void)n_in; (void)out_size;

    gat_setup<<<1, 256, 0, stream>>>(W, We, att_src, att_dst, att_edge, WTa, w_e_att);
    gat_init<<<(NN * 32 + 255) / 256, 256, 0, stream>>>(out, bias, ea_dot, cnt, amax, denom);
    gat_gemm<<<((NN / 16) + 7) / 8, 256, 0, stream>>>(x, WTa, h, a_src, a_dst);
    gat_edgeA<<<(NE + 255) / 256, 256, 0, stream>>>(eidx, eattr, w_e_att, alpha, ea_dot, cnt);
    gat_edgeB<<<(NT + 255) / 256, 256, 0, stream>>>(eidx, a_src, a_dst, ea_dot, cnt, alpha, amax);
    gat_edgeC<<<(NT + 255) / 256, 256, 0, stream>>>(eidx, amax, alpha, denom);
    gat_edgeD<<<(NT + 7) / 8, 256, 0, stream>>>(eidx, alpha, denom, h, out);
}